// HdcGenericEncoder_71279277245090
// MI455X (gfx1250) — compile-verified
//
#include <hip/hip_runtime.h>
#include <math.h>

#define T_STEPS 2048
#define NCH 4
#define LEVELS 100
#define MFCC_LEVELS 1000
#define HV_D 10000
#define TILE_D 128
#define TPAD (TILE_D + 2)

typedef float v2f  __attribute__((ext_vector_type(2)));
typedef float v8f  __attribute__((ext_vector_type(8)));
typedef unsigned int u32x4 __attribute__((ext_vector_type(4)));
typedef int   i32x4 __attribute__((ext_vector_type(4)));
typedef int   i32x8 __attribute__((ext_vector_type(8)));

__device__ __constant__ int MFCC_IDX[100] = {
    14, 15, 16, 17, 18, 20, 22, 24, 26, 27, 28, 29, 31, 37, 38, 39,
    105, 106, 107, 108, 109, 111, 113, 115, 117, 118, 120, 128, 129, 130,
    196, 197, 198, 199, 200, 201, 202, 203, 204, 206, 208, 209, 210, 211,
    212, 213, 219, 220, 221, 285, 287, 288, 289, 290, 291, 293, 295, 297,
    299, 300, 301, 302, 310, 311, 312, 374, 376, 378, 379, 380, 381, 382,
    384, 386, 388, 390, 391, 392, 393, 395, 401, 402, 403, 467, 469, 470,
    471, 472, 473, 474, 475, 477, 479, 481, 482, 483, 484, 486, 492, 494};

// fast tanh via native exp: tanh(x) = (e^{2x}-1)/(e^{2x}+1)
__device__ __forceinline__ float fast_tanhf(float x) {
  float e = __expf(2.f * x);
  return __fdividef(e - 1.f, e + 1.f);
}

// ---------------------------------------------------------------------------
// Kernel 1: quantize signal levels (RNE == jnp.round), MFCC norm + indices
// ---------------------------------------------------------------------------
__global__ __launch_bounds__(256) void prep_kernel(
    const float* __restrict__ signals, const float* __restrict__ feat,
    int* __restrict__ idx, int* __restrict__ midx) {
  const int tid = threadIdx.x;
  for (int i = tid; i < T_STEPS * NCH; i += blockDim.x) {
    int q = (int)rintf(signals[i] * (float)(LEVELS - 1));
    q = q < 0 ? 0 : (q > LEVELS - 1 ? LEVELS - 1 : q);
    idx[i] = q;
  }
  __shared__ float sm[128];
  __shared__ float inv_norm;
  float v = 0.f;
  if (tid < 100) v = feat[MFCC_IDX[tid]];
  if (tid < 128) sm[tid] = (tid < 100) ? v * v : 0.f;
  __syncthreads();
  if (tid == 0) {
    float s = 0.f;
    for (int i = 0; i < 100; ++i) s += sm[i];
    inv_norm = rsqrtf(s);
  }
  __syncthreads();
  if (tid < 100) {
    int q = (int)rintf(v * inv_norm * (float)(MFCC_LEVELS - 1));
    q = q < 0 ? 0 : (q > MFCC_LEVELS - 1 ? MFCC_LEVELS - 1 : q);
    midx[tid] = q;
  }
}

// ---------------------------------------------------------------------------
// Kernel 2: Sinusoid projections via V_WMMA_F32_16X16X4_F32 (K=3 padded to 4)
// A = x[s] broadcast in every M row; B = per-column weights; every D row of
// the 16x16 result equals proj, so VGPR0 of D gives proj for this lane's n.
// Branchless operand fill (unconditional loads + v_cndmask), native sin/cos.
// ---------------------------------------------------------------------------
__global__ __launch_bounds__(256) void sin_combo_kernel(
    const float* __restrict__ feat, const float* __restrict__ sin_w,
    const float* __restrict__ sin_b, float* __restrict__ combo) {
  const int lane = threadIdx.x & 31;
  const int wave = threadIdx.x >> 5;
  const int d0 = (blockIdx.x * 8 + wave) * 16;
  const int n = lane & 15;
  const int dreal = d0 + n;
  const int d = dreal < HV_D ? dreal : HV_D - 1;  // clamp loads, guard store
  const bool lo16 = (lane < 16);
  const int kA = lo16 ? 0 : 2;  // K-slot pair owned by this lane

  const int SLIST[9] = {0, 15, 17, 3, 4, 5, 6, 11, 12};
  const int GRP[9]   = {0, 0,  0,  1, 1, 1, 2, 3,  3};
  float g[4] = {0.f, 0.f, 0.f, 0.f};

#pragma unroll
  for (int si = 0; si < 9; ++si) {
    const int s = SLIST[si];
    // unconditional, always in-bounds loads; lane-half pick via select
    const float x_k  = feat[546 + s * 3 + kA];
    const float x_1  = feat[546 + s * 3 + 1];
    const float* wp  = sin_w + ((size_t)s * HV_D + d) * 3;
    const float w_k  = wp[kA];
    const float w_1  = wp[1];
    v2f a;
    a.x = x_k;
    a.y = lo16 ? x_1 : 0.f;  // K=3 -> K3 padded with 0
    v2f b;
    b.x = w_k;
    b.y = lo16 ? w_1 : 0.f;
    v8f c = {};
    v8f dm = __builtin_amdgcn_wmma_f32_16x16x4_f32(
        false, a, false, b, (short)0, c, false, false);
    float proj = dm[0];
    // args are O(10) -> native v_sin/v_cos path is accurate and cheap
    float fh = __cosf(proj + sin_b[(size_t)s * HV_D + d]) * __sinf(proj);
    g[GRP[si]] += fh;
  }
  float cv = g[0] * g[1] * g[2] * g[3];
  if (lo16 && dreal < HV_D) combo[dreal] = cv;
}

// ---------------------------------------------------------------------------
// Kernel 3: main HDC bind/bundle + 3-gram. One block = 128 D-columns.
// level_weight tile (100 x 128) staged into LDS by the Tensor Data Mover
// (pad_enable gives the 130-float row pitch; tensor_dim0 = D - d0 makes
// right-edge OOB reads return zero). 2-column wrap halo loaded manually.
// ---------------------------------------------------------------------------
__global__ __launch_bounds__(TILE_D) void ngram_kernel(
    const float* __restrict__ keys_w, const float* __restrict__ level_w,
    const int* __restrict__ idx, float* __restrict__ sample) {
  __shared__ float s_keys[NCH * TPAD];
  __shared__ float s_lw[LEVELS * TPAD];

  const int tid = threadIdx.x;
  const int d0 = blockIdx.x * TILE_D;

  // wrap halo: LDS columns 0,1 = global columns (d0-2), (d0-1) mod D
  for (int i = tid; i < (NCH + LEVELS) * 2; i += blockDim.x) {
    int row = i >> 1, col = i & 1;
    int gcol = d0 - 2 + col;
    if (gcol < 0) gcol += HV_D;
    if (row < NCH)
      s_keys[row * TPAD + col] = keys_w[(size_t)row * HV_D + gcol];
    else
      s_lw[(row - NCH) * TPAD + col] = level_w[(size_t)(row - NCH) * HV_D + gcol];
  }
  // keys main body (tiny) via plain loads
  for (int i = tid; i < NCH * TILE_D; i += blockDim.x) {
    int c = i / TILE_D, x = i - c * TILE_D, gcol = d0 + x;
    s_keys[c * TPAD + 2 + x] =
        (gcol < HV_D) ? keys_w[(size_t)c * HV_D + gcol] : 0.f;
  }

  // TDM descriptor load of the 100x128 level_weight tile (wave 0 issues)
  if (tid < 32) {
    unsigned lds_base = (unsigned)(uintptr_t)(&s_lw[0]) + 8u;  // col offset 2
    unsigned long long ga = (unsigned long long)(uintptr_t)(level_w + d0);
    unsigned tdim0 = (unsigned)(HV_D - d0);  // remaining cols -> OOB reads 0

    u32x4 g0;
    g0[0] = 1u;                                   // count=1, user descriptor
    g0[1] = lds_base;                             // lds_addr
    g0[2] = (unsigned)(ga & 0xFFFFFFFFull);       // global_addr lo
    g0[3] = (unsigned)((ga >> 32) & 0x1FFFFFFull) // global_addr hi
            | (2u << 30);                         // type = 2 ("image")

    i32x8 g1;
    g1[0] = (int)((2u << 16)      // data_size = 4 bytes
                  | (1u << 20)    // pad_enable
                  | (6u << 22)    // pad_interval: 128 DWORDs
                  | (1u << 25));  // pad_amount: 2 DWORDs -> 130-float pitch
    g1[1] = (int)((tdim0 & 0xFFFFu) << 16);                        // tensor_dim0 lo
    g1[2] = (int)(((tdim0 >> 16) & 0xFFFFu) | ((unsigned)LEVELS << 16)); // dim0 hi | tensor_dim1
    g1[3] = (int)((unsigned)TILE_D << 16);   // tensor_dim1 hi=0 | tile_dim0=128
    g1[4] = (int)LEVELS;                     // tile_dim1=100 | tile_dim2=0
    g1[5] = (int)HV_D;                       // tensor_dim0_stride = D
    g1[6] = 0;
    g1[7] = 0;

    i32x4 gz  = {0, 0, 0, 0};                // 2D tensor: groups 2/3 unused
    i32x8 gz8 = {0, 0, 0, 0, 0, 0, 0, 0};    // 6-arg form's extra group
    __builtin_amdgcn_tensor_load_to_lds(g0, g1, gz, gz, gz8, 0);
    __builtin_amdgcn_s_wait_tensorcnt(0);
  }
  __syncthreads();

  // thread j produces sample_hv[d0+j] = sum_w p(w,d-2)*p(w+1,d-1)*p(w+2,d)
  const int j = tid;
  const float k0a = s_keys[0 * TPAD + j], k0b = s_keys[0 * TPAD + j + 1],
              k0c = s_keys[0 * TPAD + j + 2];
  const float k1a = s_keys[1 * TPAD + j], k1b = s_keys[1 * TPAD + j + 1],
              k1c = s_keys[1 * TPAD + j + 2];
  const float k2a = s_keys[2 * TPAD + j], k2b = s_keys[2 * TPAD + j + 1],
              k2c = s_keys[2 * TPAD + j + 2];
  const float k3a = s_keys[3 * TPAD + j], k3b = s_keys[3 * TPAD + j + 1],
              k3c = s_keys[3 * TPAD + j + 2];

  const int4* iv4 = (const int4*)idx;  // 4 level indices per t, 16B aligned
  float acc = 0.f, aD2 = 0.f, aD1 = 0.f, bD1 = 0.f;
  for (int t = 0; t < T_STEPS; ++t) {
    const int4 iv = iv4[t];  // uniform -> scalar load
    const float* r0 = &s_lw[iv.x * TPAD + j];
    const float* r1 = &s_lw[iv.y * TPAD + j];
    const float* r2 = &s_lw[iv.z * TPAD + j];
    const float* r3 = &s_lw[iv.w * TPAD + j];
    float p0 = k0a * r0[0] + k1a * r1[0] + k2a * r2[0] + k3a * r3[0];
    float p1 = k0b * r0[1] + k1b * r1[1] + k2b * r2[1] + k3b * r3[1];
    float p2 = k0c * r0[2] + k1c * r1[2] + k2c * r2[2] + k3c * r3[2];
    if (t >= 2) acc += aD2 * bD1 * p2;  // window w = t-2
    aD2 = aD1;
    aD1 = p0;
    bD1 = p1;
  }
  const int d = d0 + j;
  if (d < HV_D) sample[d] = acc;
}

// ---------------------------------------------------------------------------
// Kernel 4: MFCC multibind (coalesced row reads), combine, hard quantize
// ---------------------------------------------------------------------------
__global__ __launch_bounds__(256) void finalize_kernel(
    const float* __restrict__ mfcc_lw, const int* __restrict__ midx,
    const float* __restrict__ sample, const float* __restrict__ combo,
    float* __restrict__ out) {
  const int d = blockIdx.x * blockDim.x + threadIdx.x;
  if (d >= HV_D) return;
  float prod = 1.f;
#pragma unroll 4
  for (int i = 0; i < 100; ++i) {
    prod *= fast_tanhf(mfcc_lw[(size_t)midx[i] * HV_D + d]);
  }
  const float s = sample[d];
  const float o = s * combo[d] * (s + prod);
  out[d] = (o > 0.f) ? 1.f : -1.f;
}

// ---------------------------------------------------------------------------
extern "C" void kernel_launch(void* const* d_in, const int* in_sizes, int n_in,
                              void* d_out, int out_size, void* d_ws,
                              size_t ws_size, hipStream_t stream) {
  const float* signals = (const float*)d_in[0];
  const float* feat    = (const float*)d_in[1];
  const float* keys_w  = (const float*)d_in[2];
  const float* level_w = (const float*)d_in[3];
  const float* sin_w   = (const float*)d_in[4];
  const float* sin_b   = (const float*)d_in[5];
  const float* mfcc_lw = (const float*)d_in[6];
  float* out = (float*)d_out;

  char* ws = (char*)d_ws;
  int*   idx    = (int*)(ws + 0);        // 8192 ints  (32 KB)
  int*   midx   = (int*)(ws + 32768);    // 100 ints
  float* combo  = (float*)(ws + 36864);  // 10000 floats
  float* sample = (float*)(ws + 81920);  // 10000 floats

  prep_kernel<<<1, 256, 0, stream>>>(signals, feat, idx, midx);
  sin_combo_kernel<<<(HV_D + 127) / 128, 256, 0, stream>>>(feat, sin_w, sin_b,
                                                           combo);
  ngram_kernel<<<(HV_D + TILE_D - 1) / TILE_D, TILE_D, 0, stream>>>(
      keys_w, level_w, idx, sample);
  finalize_kernel<<<(HV_D + 255) / 256, 256, 0, stream>>>(mfcc_lw, midx,
                                                          sample, combo, out);
}